// VAR1_38654705664214
// MI455X (gfx1250) — compile-verified
//
#include <hip/hip_runtime.h>

// VAR(1) scan: z[:,0] = eps[:,0]*exp(0.5*logvar); z[:,t] = z[:,t-1] @ A^T + eps[:,t]
// B=64, S=4096, D=128, fp32. Chunked two-pass parallel scan with fp32 WMMA.
// Memory floor ~268MB/23.3TB/s ~= 11.5us; two-pass scheme ~400MB ~= 17us with
// 256 blocks x 8 waves of parallelism instead of 4 blocks for the naive scan.

typedef __attribute__((ext_vector_type(2))) float v2f;
typedef __attribute__((ext_vector_type(8))) float v8f;

#define BB   64
#define SS   4096
#define DD   128
#define LCH  64     // chunk length
#define NCH  64     // number of chunks (SS / LCH)
#define LDSP 130    // padded LDS row stride (floats) -> conflict-free banks

__device__ __forceinline__ v8f wmma_f32(v2f a, v2f b, v8f c) {
  // V_WMMA_F32_16X16X4_F32: D = A(16x4) x B(4x16) + C(16x16), all fp32
  return __builtin_amdgcn_wmma_f32_16x16x4_f32(
      false, a, false, b, (short)0, c, false, false);
}

// ---------------------------------------------------------------------------
// Pass 1: per-(chunk, 16-row batch tile) local recurrence with zero carry-in.
// grid = (NCH, 4), block = 256 (8 waves; wave w owns output columns [16w,16w+16))
// ---------------------------------------------------------------------------
__global__ void var1_pass1(const float* __restrict__ eps,
                           const float* __restrict__ Amat,
                           const float* __restrict__ logvar,
                           float* __restrict__ zout) {
  __shared__ float zbuf[16 * LDSP];

  const int c    = blockIdx.x;
  const int b0   = blockIdx.y * 16;
  const int tid  = threadIdx.x;
  const int wave = tid >> 5;
  const int lane = tid & 31;
  const int half = lane >> 4;     // 0: lanes 0-15, 1: lanes 16-31
  const int l15  = lane & 15;
  const int n0   = wave * 16;
  const int t0   = c * LCH;

  // Preload constant B fragments: B[k][n] = A[n][k]
  v2f bfrag[32];
#pragma unroll
  for (int i = 0; i < 32; ++i) {
    const float* p = Amat + (n0 + l15) * DD + (4 * i + 2 * half);
    bfrag[i] = *(const v2f*)p;
  }

  // Prologue: z_local at t0 (chunk 0 applies the sqrt(exp(logvar)) init scale)
  for (int it = 0; it < 8; ++it) {
    int j = tid + 256 * it;              // 16*128 elements
    int r = j >> 7, d = j & 127;
    float v = eps[(size_t)(b0 + r) * SS * DD + (size_t)t0 * DD + d];
    if (c == 0) v *= __expf(0.5f * logvar[d]);
    zbuf[r * LDSP + d] = v;
    zout[(size_t)(b0 + r) * SS * DD + (size_t)t0 * DD + d] = v;
  }
  __syncthreads();

  for (int t = t0 + 1; t < t0 + LCH; ++t) {
    // C accumulator 0 initialized with the eps tile (add comes for free)
    v8f acc0;
#pragma unroll
    for (int v = 0; v < 8; ++v) {
      int m = 8 * half + v;
      acc0[v] = eps[(size_t)(b0 + m) * SS * DD + (size_t)t * DD + (n0 + l15)];
    }
    v8f acc1 = {};

    // Preload all A fragments (z_{t-1}) -> one LDS wait, then uninterrupted WMMAs
    v2f afrag[32];
#pragma unroll
    for (int i = 0; i < 32; ++i)
      afrag[i] = *(const v2f*)&zbuf[l15 * LDSP + 4 * i + 2 * half];

    // Two independent K-chains (even/odd K-tiles) to halve dependent latency
#pragma unroll
    for (int i = 0; i < 16; ++i) {
      acc0 = wmma_f32(afrag[2 * i],     bfrag[2 * i],     acc0);
      acc1 = wmma_f32(afrag[2 * i + 1], bfrag[2 * i + 1], acc1);
    }
    __syncthreads();   // all waves finished reading z_{t-1}
#pragma unroll
    for (int v = 0; v < 8; ++v) {
      int m = 8 * half + v;
      float x = acc0[v] + acc1[v];
      zbuf[m * LDSP + n0 + l15] = x;
      zout[(size_t)(b0 + m) * SS * DD + (size_t)t * DD + (n0 + l15)] = x;
    }
    __syncthreads();
  }
}

// ---------------------------------------------------------------------------
// A^64 by 6 WMMA squarings, ping-pong in d_ws. grid=(1), block=256.
// Result lands in ws + DD*DD (buf1).
// ---------------------------------------------------------------------------
__global__ void var1_matpow(const float* __restrict__ Amat,
                            float* __restrict__ ws) {
  float* buf0 = ws;
  float* buf1 = ws + DD * DD;
  const int tid  = threadIdx.x;
  const int wave = tid >> 5;
  const int lane = tid & 31;
  const int half = lane >> 4;
  const int l15  = lane & 15;

  const float* src = Amat;
  float* dst = buf0;
  for (int itr = 0; itr < 6; ++itr) {        // A^2,4,8,16,32,64
    for (int tt = 0; tt < 8; ++tt) {
      int tile = wave + 8 * tt;              // 64 16x16 output tiles
      int mt = tile >> 3, nt = tile & 7;
      v8f acc = {};
#pragma unroll
      for (int i = 0; i < 32; ++i) {
        int k0 = 4 * i + 2 * half;
        v2f a = *(const v2f*)&src[(mt * 16 + l15) * DD + k0];
        v2f b;
        b.x = src[(k0)     * DD + nt * 16 + l15];
        b.y = src[(k0 + 1) * DD + nt * 16 + l15];
        acc = wmma_f32(a, b, acc);
      }
#pragma unroll
      for (int v = 0; v < 8; ++v)
        dst[(mt * 16 + 8 * half + v) * DD + nt * 16 + l15] = acc[v];
    }
    __threadfence();
    __syncthreads();
    src = dst;
    dst = (dst == buf0) ? buf1 : buf0;
  }
}

// ---------------------------------------------------------------------------
// Sequential carry combine: v_c = v_{c-1} @ (A^64)^T + z_local_end(c).
// Stores every v_c ([64][64][128]) into carr. grid=(1), block=256.
// ---------------------------------------------------------------------------
__global__ void var1_carries(const float* __restrict__ zl,
                             const float* __restrict__ MP,   // A^64
                             float* __restrict__ carr) {
  __shared__ float vbuf[BB * LDSP];
  const int tid  = threadIdx.x;
  const int wave = tid >> 5;
  const int lane = tid & 31;
  const int half = lane >> 4;
  const int l15  = lane & 15;
  const int n0   = wave * 16;

  // v_0 = z_local at t = LCH-1 (chunk 0 is already true)
  for (int it = 0; it < 32; ++it) {
    int j = tid + 256 * it;                  // 64*128 elements
    int b = j >> 7, d = j & 127;
    float v = zl[(size_t)b * SS * DD + (size_t)(LCH - 1) * DD + d];
    vbuf[b * LDSP + d] = v;
    carr[b * DD + d] = v;
  }
  __syncthreads();

  v2f bfrag[32];                             // B[k][n] = (A^64)[n][k]
#pragma unroll
  for (int i = 0; i < 32; ++i)
    bfrag[i] = *(const v2f*)&MP[(n0 + l15) * DD + 4 * i + 2 * half];

  for (int c = 1; c < NCH; ++c) {
    const size_t tlast = (size_t)(c * LCH + LCH - 1) * DD;
    v8f acc[4];
#pragma unroll
    for (int mt = 0; mt < 4; ++mt) {
#pragma unroll
      for (int v = 0; v < 8; ++v) {
        int m = mt * 16 + 8 * half + v;
        acc[mt][v] = zl[(size_t)m * SS * DD + tlast + n0 + l15];
      }
#pragma unroll
      for (int i = 0; i < 32; ++i) {
        v2f a = *(const v2f*)&vbuf[(mt * 16 + l15) * LDSP + 4 * i + 2 * half];
        acc[mt] = wmma_f32(a, bfrag[i], acc[mt]);
      }
    }
    __syncthreads();
#pragma unroll
    for (int mt = 0; mt < 4; ++mt)
#pragma unroll
      for (int v = 0; v < 8; ++v) {
        int m = mt * 16 + 8 * half + v;
        float x = acc[mt][v];
        vbuf[m * LDSP + n0 + l15] = x;
        carr[(size_t)c * BB * DD + m * DD + n0 + l15] = x;
      }
    __syncthreads();
  }
}

// ---------------------------------------------------------------------------
// Fixup: for chunk c>=1, w_0 = v_{c-1}; w <- w @ A^T; z[t0+j] += w.
// grid = (NCH-1, 4), block = 256.
// ---------------------------------------------------------------------------
__global__ void var1_fixup(const float* __restrict__ Amat,
                           const float* __restrict__ carr,
                           float* __restrict__ zout) {
  __shared__ float wbuf[16 * LDSP];
  const int c    = blockIdx.x + 1;
  const int b0   = blockIdx.y * 16;
  const int tid  = threadIdx.x;
  const int wave = tid >> 5;
  const int lane = tid & 31;
  const int half = lane >> 4;
  const int l15  = lane & 15;
  const int n0   = wave * 16;
  const int t0   = c * LCH;

  v2f bfrag[32];
#pragma unroll
  for (int i = 0; i < 32; ++i)
    bfrag[i] = *(const v2f*)&Amat[(n0 + l15) * DD + 4 * i + 2 * half];

  for (int it = 0; it < 8; ++it) {
    int j = tid + 256 * it;
    int r = j >> 7, d = j & 127;
    wbuf[r * LDSP + d] =
        carr[(size_t)(c - 1) * BB * DD + (size_t)(b0 + r) * DD + d];
  }
  __syncthreads();

  for (int j = 0; j < LCH; ++j) {
    const int t = t0 + j;
    v8f acc0 = {};
    v8f acc1 = {};
    v2f afrag[32];
#pragma unroll
    for (int i = 0; i < 32; ++i)
      afrag[i] = *(const v2f*)&wbuf[l15 * LDSP + 4 * i + 2 * half];
#pragma unroll
    for (int i = 0; i < 16; ++i) {
      acc0 = wmma_f32(afrag[2 * i],     bfrag[2 * i],     acc0);
      acc1 = wmma_f32(afrag[2 * i + 1], bfrag[2 * i + 1], acc1);
    }
    // z_true = z_local + w_{j+1}
#pragma unroll
    for (int v = 0; v < 8; ++v) {
      int m = 8 * half + v;
      size_t idx = (size_t)(b0 + m) * SS * DD + (size_t)t * DD + (n0 + l15);
      zout[idx] += acc0[v] + acc1[v];
    }
    __syncthreads();
#pragma unroll
    for (int v = 0; v < 8; ++v)
      wbuf[(8 * half + v) * LDSP + n0 + l15] = acc0[v] + acc1[v];
    __syncthreads();
  }
}

// ---------------------------------------------------------------------------
extern "C" void kernel_launch(void* const* d_in, const int* in_sizes, int n_in,
                              void* d_out, int out_size, void* d_ws, size_t ws_size,
                              hipStream_t stream) {
  const float* eps = (const float*)d_in[0];   // [64,4096,128] f32
  const float* A   = (const float*)d_in[1];   // [128,128] f32
  const float* lv  = (const float*)d_in[2];   // [128] f32
  float* z  = (float*)d_out;                  // [64,4096,128] f32
  float* ws = (float*)d_ws;

  float* MP   = ws + DD * DD;       // A^64 (buf1 of ping-pong)
  float* carr = ws + 2 * DD * DD;   // [NCH][64][128] chunk-end true states (2 MB)

  var1_pass1  <<<dim3(NCH, 4),     256, 0, stream>>>(eps, A, lv, z);
  var1_matpow <<<dim3(1),          256, 0, stream>>>(A, ws);
  var1_carries<<<dim3(1),          256, 0, stream>>>(z, MP, carr);
  var1_fixup  <<<dim3(NCH - 1, 4), 256, 0, stream>>>(A, carr, z);
}